// Additive_Attention_34943853920930
// MI455X (gfx1250) — compile-verified
//
#include <hip/hip_runtime.h>
#include <hip/hip_bf16.h>

// Problem constants (fixed by the reference)
#define AB 16   // batch
#define AQ 64   // queries
#define AK 512  // keys
#define AH 256  // hidden == D

typedef __attribute__((ext_vector_type(2))) float    v2f;
typedef __attribute__((ext_vector_type(8))) float    v8f;
typedef __attribute__((ext_vector_type(4))) unsigned v4u;
typedef __attribute__((ext_vector_type(8))) int      v8i;
typedef __attribute__((ext_vector_type(4))) int      v4i;

// Native CDNA5 v_tanh_f32 if exposed, else libm tanh.
#if defined(__has_builtin)
#  if __has_builtin(__builtin_amdgcn_tanhf)
#    define TANHF(x) __builtin_amdgcn_tanhf(x)
#  elif __has_builtin(__builtin_amdgcn_tanh_f32)
#    define TANHF(x) __builtin_amdgcn_tanh_f32(x)
#  else
#    define TANHF(x) tanhf(x)
#  endif
#else
#  define TANHF(x) tanhf(x)
#endif

// Tensor Data Mover (this toolchain: 6-arg clang-23 form:
// (v4u g0, v8i g1, v4i g2, v4i g3, v8i g4, i32 cpol)).
#if defined(__has_builtin)
#  if __has_builtin(__builtin_amdgcn_tensor_load_to_lds) && \
      __has_builtin(__builtin_amdgcn_s_wait_tensorcnt)
#    define HAVE_TDM 1
#  endif
#endif

// LDS B-panel row stride (floats). 72 => upper/lower half-wave rows land
// 16 banks apart (2*72 mod 64 == 16): conflict-free dual-row reads.
#define BROW 72
#define KCHUNK 64
#define NPANEL 64
#define CHUNKF (KCHUNK * BROW)   // floats per LDS buffer

#if defined(HAVE_TDM)
// Issue one TDM descriptor: load a KCHUNK x NPANEL f32 tile (row stride N)
// from gsrc into LDS at ldst, padding 8 DWORDs every 64 DWORDs so rows land
// at stride BROW. Caller is responsible for s_wait_tensorcnt + barrier.
__device__ __forceinline__ void tdm_stage_panel(const float* gsrc,
                                                const float* ldst, int N)
{
  const unsigned long long gaddr = (unsigned long long)(const void*)gsrc;
  const unsigned lds_off = (unsigned)(unsigned long long)(const void*)ldst;
  v4u g0;
  g0[0] = 1u;                                        // count=1, user mode
  g0[1] = lds_off;                                   // lds_addr (bytes)
  g0[2] = (unsigned)(gaddr & 0xFFFFFFFFull);         // global_addr[31:0]
  g0[3] = (unsigned)((gaddr >> 32) & 0x1FFFFFFull)   // global_addr[56:32]
          | (2u << 30);                              // type = 2 ("image")
  // data_size=2 (4B) @[17:16]; pad_enable @20; pad_interval=5 (64 DW)
  // @[24:22]; pad_amount=7 (8 DW) @[31:25]  => LDS row stride 72 floats.
  const unsigned td0 = (unsigned)N;                  // tensor_dim0
  const unsigned td1 = KCHUNK;                       // tensor_dim1
  const unsigned long long str0 = (unsigned long long)N;
  v8i g1;
  g1[0] = (int)((2u << 16) | (1u << 20) | (5u << 22) | (7u << 25));
  g1[1] = (int)((td0 & 0xFFFFu) << 16);              // [63:48]=td0 lo16
  g1[2] = (int)((td0 >> 16) | ((td1 & 0xFFFFu) << 16));
  g1[3] = (int)((td1 >> 16) | ((unsigned)NPANEL << 16)); // tile_dim0
  g1[4] = (int)KCHUNK;                               // tile_dim1; tile_dim2=0
  g1[5] = (int)(str0 & 0xFFFFFFFFull);
  g1[6] = (int)((str0 >> 32) & 0xFFFFull);           // stride1 = 0
  g1[7] = 0;
  v4i gz  = {0, 0, 0, 0};                            // 2D: groups 2/3 NULL
  v8i gz8 = {0, 0, 0, 0, 0, 0, 0, 0};                // unused extra group
  __builtin_amdgcn_tensor_load_to_lds(g0, g1, gz, gz, gz8, 0);
}
#endif

// ---------------------------------------------------------------------------
// Batched row-major GEMM: C[b] = A[b](MxK) * B[b](KxN), f32, via
// V_WMMA_F32_16X16X4_F32. Block = 128 threads (4 waves). Block tile 64x64:
// wave w owns rows m0+16w, all four 16-col n-tiles (4 accumulators).
// B k-panels are double-buffered in LDS: the Tensor Data Mover prefetches
// chunk c+1 while all waves compute chunk c (one barrier per chunk). A
// fragments for a whole chunk are preloaded as one 16x global_load_b64
// clause so WMMAs never stall on a just-issued load.
// Requires M%64==0, N%64==0, K%64==0.
// ---------------------------------------------------------------------------
__global__ __launch_bounds__(128) void wmma_gemm_f32(
    const float* __restrict__ A, const float* __restrict__ Bm,
    float* __restrict__ C, int M, int N, int K,
    long sA, long sB, long sC)
{
  __shared__ float lds_b[2 * CHUNKF];

  const int lane = threadIdx.x & 31;
  const int wave = threadIdx.x >> 5;          // 0..3 -> m sub-tile
  const int half = lane >> 4;                 // 0 or 1
  const int r    = lane & 15;

  const int m0 = blockIdx.x * 64 + wave * 16;
  const int nb = blockIdx.y * NPANEL;
  const int b  = blockIdx.z;

  const float* Ab = A  + (long)b * sA;
  const float* Bb = Bm + (long)b * sB;
  float*       Cb = C  + (long)b * sC;

  v8f acc[4] = {};

  // This lane's A-row slot (ISA 16x4 f32 A layout: lanes 0-15 hold K=0,1;
  // lanes 16-31 hold K=2,3).
  const float* arow = Ab + (long)(m0 + r) * K + 2 * half;

  const int nchunks = K / KCHUNK;

  auto compute_chunk = [&](int c) {
    const float* buf   = &lds_b[(c & 1) * CHUNKF];
    const float* achnk = arow + c * KCHUNK;
    // Preload all A fragments for this chunk as one load clause.
    v2f afrag[KCHUNK / 4];
#pragma unroll
    for (int i = 0; i < KCHUNK / 4; ++i)
      afrag[i] = *(const v2f*)(achnk + 4 * i);
#pragma unroll
    for (int kk = 0; kk < KCHUNK; kk += 4) {
      const v2f a = afrag[kk >> 2];
      const float* bbase = &buf[(kk + 2 * half) * BROW + r];
#pragma unroll
      for (int nt = 0; nt < 4; ++nt) {
        v2f bf;
        bf[0] = bbase[nt * 16];            // B[kk+2h  ][nb+16nt+r]
        bf[1] = bbase[nt * 16 + BROW];     // B[kk+2h+1][nb+16nt+r]
        acc[nt] = __builtin_amdgcn_wmma_f32_16x16x4_f32(
            false, a, false, bf, (short)0, acc[nt], false, false);
      }
    }
  };

#if defined(HAVE_TDM)
  // Software pipeline: TDM prefetch of chunk c+1 overlaps compute of chunk c.
  if (threadIdx.x < 32)
    tdm_stage_panel(Bb + nb, &lds_b[0], N);
  for (int c = 0; c < nchunks; ++c) {
    if (threadIdx.x < 32)
      __builtin_amdgcn_s_wait_tensorcnt(0);   // chunk c has landed
    __syncthreads();                          // publish; buf[(c+1)&1] is free
    if ((c + 1 < nchunks) && threadIdx.x < 32)
      tdm_stage_panel(Bb + (long)(c + 1) * KCHUNK * N + nb,
                      &lds_b[((c + 1) & 1) * CHUNKF], N);
    compute_chunk(c);
  }
#else
  for (int c = 0; c < nchunks; ++c) {
    __syncthreads();
    float* buf = &lds_b[(c & 1) * CHUNKF];
    const float* src = Bb + (long)c * KCHUNK * N + nb;
    for (int f = threadIdx.x; f < KCHUNK * (NPANEL / 4); f += 128) {
      const int kr = f >> 4;        // row in chunk
      const int c4 = f & 15;        // float4 within row
      *(float4*)&buf[kr * BROW + c4 * 4] =
          *(const float4*)&src[(long)kr * N + c4 * 4];
    }
    __syncthreads();
    compute_chunk(c);
  }
#endif

  // C/D layout: VGPR j -> row m0+j (lanes 0-15) / m0+8+j (lanes 16-31)
#pragma unroll
  for (int nt = 0; nt < 4; ++nt) {
    float* cp = Cb + (long)(m0 + 8 * half) * N + nb + nt * 16 + r;
#pragma unroll
    for (int j = 0; j < 8; ++j)
      cp[(long)j * N] = acc[nt][j];
  }
}

// ---------------------------------------------------------------------------
// Fused scores + mask + softmax:
//   s[b,q,k] = sum_h w_v[h] * tanh(qproj[b,q,h] + kproj[b,k,h])
//   s = (k < valid_len[b]) ? s : -1e6 ; attn = softmax_k(s)
// One block per (b,q); 512 threads = one key each. Avoids materializing the
// 537 MB (B,Q,K,H) features tensor. VALU/TRANS bound (134M v_tanh_f32 total).
// ---------------------------------------------------------------------------
__global__ __launch_bounds__(512) void scores_softmax(
    const float* __restrict__ qproj, const float* __restrict__ kproj,
    const float* __restrict__ wv, const int* __restrict__ valid_lens,
    float* __restrict__ attn)
{
  __shared__ float lq[AH];
  __shared__ float lw[AH];
  __shared__ float lr[AK];

  const int tid = threadIdx.x;     // key index
  const int bq  = blockIdx.x;      // flattened (b,q)
  const int b   = bq >> 6;         // bq / AQ

  for (int i = tid; i < AH; i += AK) {
    lq[i] = qproj[(long)bq * AH + i];
    lw[i] = wv[i];
  }
  __syncthreads();

  float s = 0.f;
  const float* kr = kproj + ((long)b * AK + tid) * AH;
#pragma unroll 4
  for (int h = 0; h < AH; h += 4) {
    float4 kv = *(const float4*)(kr + h);
    s = fmaf(lw[h + 0], TANHF(lq[h + 0] + kv.x), s);
    s = fmaf(lw[h + 1], TANHF(lq[h + 1] + kv.y), s);
    s = fmaf(lw[h + 2], TANHF(lq[h + 2] + kv.z), s);
    s = fmaf(lw[h + 3], TANHF(lq[h + 3] + kv.w), s);
  }

  if (tid >= valid_lens[b]) s = -1e6f;   // reference NEG_INF fill

  // max over K
  lr[tid] = s;
  __syncthreads();
  for (int off = AK / 2; off > 0; off >>= 1) {
    if (tid < off) lr[tid] = fmaxf(lr[tid], lr[tid + off]);
    __syncthreads();
  }
  const float mx = lr[0];
  __syncthreads();

  // sum of exp
  const float e = __expf(s - mx);
  lr[tid] = e;
  __syncthreads();
  for (int off = AK / 2; off > 0; off >>= 1) {
    if (tid < off) lr[tid] += lr[tid + off];
    __syncthreads();
  }
  const float inv = 1.0f / lr[0];

  attn[(long)bq * AK + tid] = e * inv;
}

// ---------------------------------------------------------------------------
// Launch: two projection GEMMs, fused scores/softmax, batched attn@values.
// d_ws: [qproj 1MB | kproj 8MB | attn 2MB]
// ---------------------------------------------------------------------------
extern "C" void kernel_launch(void* const* d_in, const int* in_sizes, int n_in,
                              void* d_out, int out_size, void* d_ws, size_t ws_size,
                              hipStream_t stream)
{
  const float* querys = (const float*)d_in[0];  // (16,64,256)
  const float* keys   = (const float*)d_in[1];  // (16,512,256)
  const float* values = (const float*)d_in[2];  // (16,512,256)
  const int*   vlen   = (const int*)d_in[3];    // (16,)
  const float* Wq     = (const float*)d_in[4];  // (256,256)
  const float* Wk     = (const float*)d_in[5];  // (256,256)
  const float* wv     = (const float*)d_in[6];  // (256,)
  float* out = (float*)d_out;                   // (16,64,256)

  float* qproj = (float*)d_ws;                           // 16*64*256
  float* kproj = qproj + (size_t)AB * AQ * AH;           // 16*512*256
  float* attn  = kproj + (size_t)AB * AK * AH;           // 16*64*512

  // q = querys @ Wq : M=1024, N=256, K=256
  wmma_gemm_f32<<<dim3((AB * AQ) / 64, AH / NPANEL, 1), 128, 0, stream>>>(
      querys, Wq, qproj, AB * AQ, AH, AH, 0, 0, 0);

  // k = keys @ Wk : M=8192, N=256, K=256
  wmma_gemm_f32<<<dim3((AB * AK) / 64, AH / NPANEL, 1), 128, 0, stream>>>(
      keys, Wk, kproj, AB * AK, AH, AH, 0, 0, 0);

  // fused scores + mask + softmax : one block per (b,q)
  scores_softmax<<<dim3(AB * AQ), 512, 0, stream>>>(
      qproj, kproj, wv, vlen, attn);

  // out = attn @ values : batched, M=64, N=256, K=512
  wmma_gemm_f32<<<dim3(AQ / 64, AH / NPANEL, AB), 128, 0, stream>>>(
      attn, values, out, AQ, AH, AK,
      (long)AQ * AK, (long)AK * AH, (long)AQ * AH);
}